// RNN_Net_67061619360064
// MI455X (gfx1250) — compile-verified
//
#include <hip/hip_runtime.h>

// Sizes from the reference
#define Bsz 64
#define Tsz 1024
#define Isz 128
#define Hsz 1024
#define Osz 64
#define DTC 0.2f

#define NUPD 64            // update blocks: one per 16-column slab of H
#define NDEC 4             // decoder blocks: one per 16-column slab of O
#define NBLK (NUPD + NDEC)
#define WSTRIDE 1028       // LDS row stride (1024 + 4) -> stride%64==4 -> conflict-free b64
#define WIH_STRIDE 132     // LDS row stride (128 + 4)
#define BH (Bsz * Hsz)

typedef float v2f __attribute__((ext_vector_type(2)));
typedef float v8f __attribute__((ext_vector_type(8)));

__device__ __forceinline__ v8f wmma_f32(v2f a, v2f b, v8f c) {
    return __builtin_amdgcn_wmma_f32_16x16x4_f32(
        false, a, false, b, (short)0, c, false, false);
}

__global__ __launch_bounds__(128, 1)
void rnn_persistent(const float* __restrict__ x,      // [B,T,I]
                    const float* __restrict__ W_ih,   // [H,I]
                    const float* __restrict__ W_hh,   // [H,H]
                    const float* __restrict__ W_out,  // [O,H]
                    float* __restrict__ out,          // [B,T,O]
                    float* __restrict__ hbase,        // [2,B,H] double buffer (zeroed)
                    unsigned* __restrict__ bar)       // zeroed
{
    __shared__ float lw [16 * WSTRIDE];     // main K=1024 weight slice (W_hh or W_out rows)
    __shared__ float lwi[16 * WIH_STRIDE];  // W_ih slice (update blocks only)

    const int  blk    = blockIdx.x;
    const bool is_dec = (blk >= NUPD);
    const int  c0     = (is_dec ? (blk - NUPD) : blk) * 16;  // output-column tile base
    const int  tid    = threadIdx.x;
    const int  lane   = tid & 31;
    const int  wave   = tid >> 5;        // 4 waves -> 4 batch tiles
    const int  m0     = wave * 16;

    // ---- one-time LDS stage of this block's 16 weight rows (K-major, padded) ----
    {
        const float* wsrc = is_dec ? (W_out + (size_t)c0 * Hsz) : (W_hh + (size_t)c0 * Hsz);
        for (int idx = tid; idx < 16 * (Hsz / 4); idx += 128) {
            const int r  = idx / (Hsz / 4);
            const int c4 = (idx % (Hsz / 4)) * 4;
            *(float4*)(lw + r * WSTRIDE + c4) = *(const float4*)(wsrc + r * Hsz + c4);
        }
        if (!is_dec) {
            for (int idx = tid; idx < 16 * (Isz / 4); idx += 128) {
                const int r  = idx / (Isz / 4);
                const int c4 = (idx % (Isz / 4)) * 4;
                *(float4*)(lwi + r * WIH_STRIDE + c4) = *(const float4*)(W_ih + (c0 + r) * Isz + c4);
            }
        }
    }
    __syncthreads();

    // Fragment geometry (V_WMMA_F32_16X16X4_F32, wave32):
    //   A (16x4): lane L holds h[m0 + (L&15)][k0 + 2*(L>>4) + {0,1}]  -> float2
    //   B (4x16): lane L holds W[c0 + (L&15)][k0 + 2*(L>>4) + {0,1}]  -> float2
    //   D: vgpr r, lane L  -> element (m0 + 8*(L>>4) + r, c0 + (L&15))
    const int n    = lane & 15;
    const int koff = (lane >> 4) * 2;
    const int arow = m0 + n;                 // A-fragment batch row
    const int drow = m0 + 8 * (lane >> 4);   // D base batch row
    const int dcol = c0 + n;

    const float* wb0  = lw  + n * WSTRIDE    + koff;
    const float* wib0 = lwi + n * WIH_STRIDE + koff;
    const float* xa0  = x + (size_t)arow * (Tsz * Isz) + koff;
    const int    haoff = arow * Hsz + koff;  // h A-frag offset within one buffer

    for (int t = 0; t < Tsz; ++t) {
        // offset-based double-buffer select keeps pointers provably GLOBAL
        const float* hc = hbase + (size_t)(t & 1) * BH;        // h_t (pre-update)
        float*       hn = hbase + (size_t)((t + 1) & 1) * BH;  // h_{t+1}

        v8f acc0 = {}, acc1 = {};

        // ---- f = h_t @ W^T  (K = 1024; B from LDS, A from L2-resident h) ----
        const float* ha = hc + haoff;
        #pragma unroll 8
        for (int k = 0; k < Hsz; k += 8) {
            acc0 = wmma_f32(*(const v2f*)(ha + k),     *(const v2f*)(wb0 + k),     acc0);
            acc1 = wmma_f32(*(const v2f*)(ha + k + 4), *(const v2f*)(wb0 + k + 4), acc1);
        }

        if (!is_dec) {
            // ---- += x[:,t,:] @ W_ih^T  (K = 128) ----
            const float* xa = xa0 + t * Isz;
            #pragma unroll
            for (int k = 0; k < Isz; k += 8) {
                acc0 = wmma_f32(*(const v2f*)(xa + k),     *(const v2f*)(wib0 + k),     acc0);
                acc1 = wmma_f32(*(const v2f*)(xa + k + 4), *(const v2f*)(wib0 + k + 4), acc1);
            }
            const v8f acc = acc0 + acc1;
            // ---- leaky update: h_{t+1} = (1-dt) h_t + dt f ----
            #pragma unroll
            for (int r = 0; r < 8; ++r) {
                const int row = drow + r;
                const float hold = hc[row * Hsz + dcol];
                hn[row * Hsz + dcol] = (1.0f - DTC) * hold + DTC * acc[r];
            }
        } else {
            const v8f acc = acc0 + acc1;
            // ---- out[:,t,:] = h_t @ W_out^T  (PRE-update state; t=0 -> zeros) ----
            #pragma unroll
            for (int r = 0; r < 8; ++r) {
                const int row = drow + r;
                out[(size_t)row * (Tsz * Osz) + t * Osz + dcol] = acc[r];
            }
        }

        // ---- device-wide barrier: release h_{t+1}/out[t], acquire for next step ----
        __threadfence();
        __syncthreads();
        if (tid == 0) {
            __hip_atomic_fetch_add(bar, 1u, __ATOMIC_ACQ_REL, __HIP_MEMORY_SCOPE_AGENT);
            const unsigned target = (unsigned)NBLK * (unsigned)(t + 1);
            while (__hip_atomic_load(bar, __ATOMIC_ACQUIRE, __HIP_MEMORY_SCOPE_AGENT) < target)
                __builtin_amdgcn_s_sleep(1);   // 64-clk poll: per-step latency matters
        }
        __syncthreads();
    }
}

extern "C" void kernel_launch(void* const* d_in, const int* in_sizes, int n_in,
                              void* d_out, int out_size, void* d_ws, size_t ws_size,
                              hipStream_t stream) {
    (void)in_sizes; (void)n_in; (void)out_size; (void)ws_size;
    const float* x     = (const float*)d_in[0];
    const float* W_ih  = (const float*)d_in[1];
    const float* W_hh  = (const float*)d_in[2];
    const float* W_out = (const float*)d_in[3];
    float* out = (float*)d_out;

    unsigned char* ws = (unsigned char*)d_ws;
    unsigned* bar = (unsigned*)ws;                   // barrier counter @ offset 0
    float* hbase = (float*)(ws + 256);               // [2,B,H] double buffer

    // Zero barrier counter + both h buffers (h_0 == 0). Graph-capture-safe,
    // re-executed on every replay -> deterministic.
    hipMemsetAsync(d_ws, 0, 256 + 2 * (size_t)BH * sizeof(float), stream);

    rnn_persistent<<<dim3(NBLK), dim3(128), 0, stream>>>(
        x, W_ih, W_hh, W_out, out, hbase, bar);
}